// BezierParameterProcessor_85478439125164
// MI455X (gfx1250) — compile-verified
//
#include <hip/hip_runtime.h>
#include <hip/hip_bf16.h>

typedef __attribute__((ext_vector_type(2))) float v2f;
typedef __attribute__((ext_vector_type(8))) float v8f;

#define HW_B   16
#define HW_N   2048
#define HW_G   4096
#define HID    128
#define CHUNKS 8          // blocks per batch in the MLP phase

#define LOG2E    1.44269504088896340736f
#define SQRT2PI  2.50662827463100050242f

// -----------------------------------------------------------------------------
// Kernel 1: layers 1-2 via V_WMMA_F32_16X16X4_F32, accumulate sum(relu(h2))
// grid = (CHUNKS, B), block = 128 threads (4 waves). Each wave owns 16-point
// tiles; h1 is staged in LDS in WMMA "A" layout source form, W2 is staged in
// LDS as K-pair float2 fragments so every B fragment is one ds_load_b64.
// -----------------------------------------------------------------------------
__global__ __launch_bounds__(128) void bezier_mlp_partial_kernel(
    const float* __restrict__ pts,   // (B, N, 2)
    const float* __restrict__ W1,    // (2, 64)
    const float* __restrict__ b1,    // (64)
    const float* __restrict__ W2,    // (64, 128)
    const float* __restrict__ b2,    // (128)
    float* __restrict__ partial)     // (B, CHUNKS, 128) sums of relu(h2)
{
    __shared__ v2f   w2pair[32 * 128];     // {W2[2k][c], W2[2k+1][c]}
    __shared__ float stage[4][16 * 66];    // per-wave 16x64 h1 tile (+pad)
    __shared__ float red[4][128];

    const int tid  = threadIdx.x;
    const int wave = tid >> 5;
    const int lane = tid & 31;
    const int half = lane >> 4;            // which 16-lane half
    const int l16  = lane & 15;
    const int chunk = blockIdx.x;
    const int b     = blockIdx.y;

    // stage W2 as K-pair fragments
    for (int j = tid; j < 32 * 128; j += 128) {
        int kp = j >> 7, c = j & 127;
        v2f w; w.x = W2[(2 * kp) * 128 + c]; w.y = W2[(2 * kp + 1) * 128 + c];
        w2pair[j] = w;
    }
    // per-lane layer-1 weights (lane owns features lane and lane+32)
    const float w1x0 = W1[lane],      w1y0 = W1[64 + lane];
    const float w1x1 = W1[32 + lane], w1y1 = W1[96 + lane];
    const float bb0  = b1[lane],      bb1  = b1[32 + lane];
    // layer-2 bias per output tile (col = nt*16 + l16)
    float bias2[8];
    #pragma unroll
    for (int nt = 0; nt < 8; ++nt) bias2[nt] = b2[nt * 16 + l16];

    __syncthreads();

    const float* p_b = pts + b * HW_N * 2;
    float* stg = stage[wave];

    float sums[8];
    #pragma unroll
    for (int nt = 0; nt < 8; ++nt) sums[nt] = 0.0f;

    const int t0 = chunk * 16;             // 16 tiles per block, 4 per wave
    for (int t = t0 + wave; t < t0 + 16; t += 4) {
        const float* pt = p_b + t * 32;
        // ---- layer 1 -> LDS stage (16 rows x 64 cols) ----
        #pragma unroll
        for (int m = 0; m < 16; ++m) {
            float x = pt[2 * m], y = pt[2 * m + 1];
            float h0 = fmaxf(x * w1x0 + y * w1y0 + bb0, 0.0f);
            float h1 = fmaxf(x * w1x1 + y * w1y1 + bb1, 0.0f);
            stg[m * 66 + lane]      = h0;
            stg[m * 66 + 32 + lane] = h1;
        }
        // stage is wave-private; LDS ops are in-order within a wave.

        // ---- layer 2: 16x128 = 8 N-tiles, K=64 in 16 steps of K=4 ----
        v8f acc[8];
        #pragma unroll
        for (int nt = 0; nt < 8; ++nt) {
            v8f c;
            #pragma unroll
            for (int v = 0; v < 8; ++v) c[v] = bias2[nt];
            acc[nt] = c;
        }
        #pragma unroll
        for (int kk = 0; kk < 16; ++kk) {
            const int kbase = 4 * kk + 2 * half;       // this lane's K pair
            v2f a = *(const v2f*)&stg[l16 * 66 + kbase];
            #pragma unroll
            for (int nt = 0; nt < 8; ++nt) {
                v2f bv = w2pair[(kbase >> 1) * 128 + nt * 16 + l16];
                acc[nt] = __builtin_amdgcn_wmma_f32_16x16x4_f32(
                    false, a, false, bv, (short)0, acc[nt], false, false);
            }
        }
        // ---- relu + per-column partial sums (8 rows live in this lane) ----
        #pragma unroll
        for (int nt = 0; nt < 8; ++nt) {
            float s = 0.0f;
            #pragma unroll
            for (int v = 0; v < 8; ++v) s += fmaxf(acc[nt][v], 0.0f);
            sums[nt] += s;
        }
    }

    // combine the two row-halves (lanes l and l+16 hold rows 0-7 / 8-15)
    #pragma unroll
    for (int nt = 0; nt < 8; ++nt) {
        float s = sums[nt] + __shfl_xor(sums[nt], 16, 32);
        if (half == 0) red[wave][nt * 16 + l16] = s;
    }
    __syncthreads();

    int col = tid;   // 0..127
    float s = red[0][col] + red[1][col] + red[2][col] + red[3][col];
    partial[(b * CHUNKS + chunk) * HID + col] = s;
}

// -----------------------------------------------------------------------------
// Kernel 2: reduce chunk partials -> mean(relu(h2)), then layer 3 matvec.
// grid = B, block = 128 (one thread per output feature). Deterministic order.
// -----------------------------------------------------------------------------
__global__ __launch_bounds__(128) void bezier_mlp_finalize_kernel(
    const float* __restrict__ partial,   // (B, CHUNKS, 128)
    const float* __restrict__ W3,        // (128, 128)
    const float* __restrict__ b3,        // (128)
    float* __restrict__ mean_out)        // (B, 128)
{
    __shared__ float mr[128];
    const int b = blockIdx.x, col = threadIdx.x;
    float s = 0.0f;
    for (int c = 0; c < CHUNKS; ++c) s += partial[(b * CHUNKS + c) * HID + col];
    mr[col] = s * (1.0f / (float)HW_N);
    __syncthreads();
    float e = b3[col];
    for (int k = 0; k < 128; ++k) e += mr[k] * W3[k * 128 + col];
    mean_out[b * HID + col] = e;
}

// -----------------------------------------------------------------------------
// Kernel 3: KDE density. grid = (G/256, B), block = 256. Per-point exponent
// coefficients pre-scaled by c2 = -0.5*log2(e)/bw^2 are staged in LDS so the
// inner loop is 1 ds_load_b128 + add + 2 fma + v_exp_f32 + acc per term.
//   arg_n(g) = c2*|p_n|^2 + c2*|g|^2 + (-2c2*p_x)*g_x + (-2c2*p_y)*g_y
// -----------------------------------------------------------------------------
__global__ __launch_bounds__(256) void bezier_density_kernel(
    const float* __restrict__ pts,       // (B, N, 2)
    const float* __restrict__ bw_ptr,    // scalar
    float* __restrict__ dens)            // (B, G)
{
    __shared__ float4 sc[HW_N];          // (A_n, Bx_n, By_n, 0)
    const int tid = threadIdx.x;
    const int b   = blockIdx.y;

    float bw = fmaxf(bw_ptr[0], 1e-6f);
    const float c2   = (-0.5f / (bw * bw)) * LOG2E;
    const float norm = 1.0f / ((float)HW_N * bw * SQRT2PI);

    const float2* p_b = (const float2*)(pts + b * HW_N * 2);
    for (int j = tid; j < HW_N; j += 256) {
        float2 p = p_b[j];
        float4 c;
        c.x = c2 * (p.x * p.x + p.y * p.y);   // A_n
        c.y = (-2.0f * c2) * p.x;             // Bx_n
        c.z = (-2.0f * c2) * p.y;             // By_n
        c.w = 0.0f;
        sc[j] = c;
    }
    __syncthreads();

    const int g = blockIdx.x * 256 + tid;
    const float gx = (float)(g & 63) * (1.0f / 63.0f);
    const float gy = (float)(g >> 6) * (1.0f / 63.0f);
    const float base = c2 * (gx * gx + gy * gy);   // loop-invariant

    float a0 = 0.f, a1 = 0.f, a2 = 0.f, a3 = 0.f;
    for (int n = 0; n < HW_N; n += 4) {
        float4 c0 = sc[n], c1 = sc[n + 1], c2v = sc[n + 2], c3 = sc[n + 3];
        float t;
        t = fmaf(c0.y, gx, fmaf(c0.z, gy, base + c0.x));
        a0 += __builtin_amdgcn_exp2f(t);
        t = fmaf(c1.y, gx, fmaf(c1.z, gy, base + c1.x));
        a1 += __builtin_amdgcn_exp2f(t);
        t = fmaf(c2v.y, gx, fmaf(c2v.z, gy, base + c2v.x));
        a2 += __builtin_amdgcn_exp2f(t);
        t = fmaf(c3.y, gx, fmaf(c3.z, gy, base + c3.x));
        a3 += __builtin_amdgcn_exp2f(t);
    }
    float d = ((a0 + a1) + (a2 + a3)) * norm;
    // sigmoid then affine to [0.01, 1.0]
    float e = __builtin_amdgcn_exp2f(-d * LOG2E);
    float sig = __builtin_amdgcn_rcpf(1.0f + e);
    dens[b * HW_G + g] = 0.01f + 0.99f * sig;
}

// -----------------------------------------------------------------------------
// Kernel 4: field_map broadcast, float4 streaming stores (memory-bound).
// -----------------------------------------------------------------------------
__global__ __launch_bounds__(256) void bezier_field_kernel(
    const float* __restrict__ mean_emb,  // (B, 128)
    float* __restrict__ out)             // (B, 128, G)
{
    const int idx = blockIdx.x * 256 + threadIdx.x;  // float4 index
    const int bh  = idx >> 10;                       // G/4 = 1024 vec4 per (b,h)
    const float v = mean_emb[bh];
    float4 val; val.x = v; val.y = v; val.z = v; val.w = v;
    ((float4*)out)[idx] = val;
}

extern "C" void kernel_launch(void* const* d_in, const int* in_sizes, int n_in,
                              void* d_out, int out_size, void* d_ws, size_t ws_size,
                              hipStream_t stream) {
    const float* pts = (const float*)d_in[0];
    const float* W1  = (const float*)d_in[1];
    const float* b1  = (const float*)d_in[2];
    const float* W2  = (const float*)d_in[3];
    const float* b2  = (const float*)d_in[4];
    const float* W3  = (const float*)d_in[5];
    const float* b3  = (const float*)d_in[6];
    const float* bw  = (const float*)d_in[7];

    float* out_density = (float*)d_out;                       // (B, 64, 64)
    float* out_field   = (float*)d_out + HW_B * HW_G;         // (B, 128, 64, 64)

    float* ws_partial = (float*)d_ws;                         // B*CHUNKS*128 floats
    float* ws_mean    = ws_partial + HW_B * CHUNKS * HID;     // B*128 floats

    bezier_mlp_partial_kernel<<<dim3(CHUNKS, HW_B), 128, 0, stream>>>(
        pts, W1, b1, W2, b2, ws_partial);
    bezier_mlp_finalize_kernel<<<HW_B, 128, 0, stream>>>(
        ws_partial, W3, b3, ws_mean);
    bezier_density_kernel<<<dim3(HW_G / 256, HW_B), 256, 0, stream>>>(
        pts, bw, out_density);
    bezier_field_kernel<<<(HW_B * HID * HW_G / 4) / 256, 256, 0, stream>>>(
        ws_mean, out_field);
}